// IMQ_24730421691123
// MI455X (gfx1250) — compile-verified
//
#include <hip/hip_runtime.h>
#include <hip/hip_bf16.h>

typedef __attribute__((ext_vector_type(2))) float v2f;
typedef __attribute__((ext_vector_type(4))) float v4f;
typedef __attribute__((ext_vector_type(8))) float v8f;

#define NPTS 1024
#define DIM  512
#define KRANK ((NPTS * NPTS - 1) / 2)   // lower median rank (0-indexed) = 524287

// ---------------- init: zero histogram + state ----------------
__global__ __launch_bounds__(256) void init_state(unsigned* hist, unsigned* state) {
    int t = threadIdx.x;
    if (t < 256) hist[t] = 0u;
    if (t == 0) { state[0] = 0u; state[1] = (unsigned)KRANK; }
}

// ---------------- row squared norms ----------------
__global__ __launch_bounds__(256) void row_sqnorm(const float* __restrict__ X,
                                                  float* __restrict__ sq) {
    int i = blockIdx.x * blockDim.x + threadIdx.x;   // one thread per row
    const float* row = X + (size_t)i * DIM;
    float acc = 0.0f;
    #pragma unroll 4
    for (int d = 0; d < DIM; d += 4) {
        v4f v = *(const v4f*)(row + d);
        acc += v.x * v.x + v.y * v.y + v.z * v.z + v.w * v.w;
    }
    sq[i] = acc;
}

// ---------------- dists via fp32 WMMA Gram matrix ----------------
// dists[i,j] = sq[i] + sq[j] - 2 * (X X^T)[i,j]
// One 16x16 tile per wave; K-loop of V_WMMA_F32_16X16X4_F32.
__global__ __launch_bounds__(256) void wmma_gram_dists(const float* __restrict__ X,
                                                       const float* __restrict__ sq,
                                                       float* __restrict__ dists) {
    int wave = blockIdx.x * 8 + (threadIdx.x >> 5);   // 4096 tiles total
    int lane = threadIdx.x & 31;
    int ti = wave >> 6;          // 64 row tiles
    int tj = wave & 63;          // 64 col tiles
    int row0 = ti * 16, col0 = tj * 16;

    int m  = lane & 15;                 // A row / B col held by this lane
    int kb = (lane >> 4) * 2;           // K sub-offset {0,2}

    const float* aptr = X + (size_t)(row0 + m) * DIM + kb;
    const float* bptr = X + (size_t)(col0 + m) * DIM + kb;

    v8f acc = {};
    #pragma unroll 8
    for (int k0 = 0; k0 < DIM; k0 += 4) {
        v2f a = *(const v2f*)(aptr + k0);
        v2f b = *(const v2f*)(bptr + k0);
        acc = __builtin_amdgcn_wmma_f32_16x16x4_f32(
            false, a, false, b, (short)0, acc, false, false);
    }

    int n  = lane & 15;                 // output column
    int hi = (lane >> 4) * 8;           // output row base
    float sqj = sq[col0 + n];
    #pragma unroll
    for (int v = 0; v < 8; ++v) {
        int i = row0 + hi + v;
        dists[(size_t)i * NPTS + col0 + n] = sq[i] + sqj - 2.0f * acc[v];
    }
}

// ---------------- radix-select median: histogram pass ----------------
__global__ __launch_bounds__(256) void hist_pass(const float* __restrict__ dists,
                                                 unsigned* __restrict__ hist,
                                                 const unsigned* __restrict__ state,
                                                 int shift, unsigned mask) {
    __shared__ unsigned lh[256];
    if (threadIdx.x < 256) lh[threadIdx.x] = 0u;
    __syncthreads();

    unsigned prefix = state[0] & mask;
    int total = NPTS * NPTS;
    int stride = gridDim.x * blockDim.x;
    for (int idx = blockIdx.x * blockDim.x + threadIdx.x; idx < total; idx += stride) {
        unsigned b = __float_as_uint(dists[idx]);
        // order-preserving transform: unsigned compare == float compare
        unsigned u = (b & 0x80000000u) ? ~b : (b | 0x80000000u);
        if ((u & mask) == prefix)
            atomicAdd(&lh[(u >> shift) & 0xFFu], 1u);
    }
    __syncthreads();
    if (threadIdx.x < 256) {
        unsigned c = lh[threadIdx.x];
        if (c) atomicAdd(&hist[threadIdx.x], c);
    }
}

// ---------------- radix-select: pick digit, update state, reset hist ----------------
__global__ void radix_select(unsigned* hist, unsigned* state, int shift) {
    unsigned krem = state[1];
    unsigned cum = 0u, digit = 255u;
    for (unsigned d = 0; d < 256u; ++d) {
        unsigned c = hist[d];
        if (cum + c > krem) { digit = d; break; }
        cum += c;
    }
    state[0] |= (digit << shift);
    state[1] = krem - cum;
    for (int i = 0; i < 256; ++i) hist[i] = 0u;
}

// ---------------- sigma = decay*median + (1-decay)*running; out sqrt(sigma) ----------
__global__ void finalize_sigma(const float* __restrict__ EMA,
                               unsigned* state, float* __restrict__ out_scalar) {
    unsigned u = state[0];
    unsigned b = (u & 0x80000000u) ? (u & 0x7FFFFFFFu) : ~u;  // inverse transform
    float med = __uint_as_float(b);
    float decay = EMA[1];
    float sigma = decay * med + (1.0f - decay) * EMA[0];
    float* sf = (float*)state;
    sf[2] = sigma;
    sf[3] = 2.0f / sigma;
    *out_scalar = sqrtf(sigma);
}

// ---------------- k = 1/sqrt(1 + dists/sigma) ----------------
__global__ __launch_bounds__(256) void compute_k(const float* __restrict__ dists,
                                                 const float* __restrict__ state_f,
                                                 float* __restrict__ kout) {
    float sigma = state_f[2];
    int total = NPTS * NPTS;
    int stride = gridDim.x * blockDim.x;
    for (int idx = blockIdx.x * blockDim.x + threadIdx.x; idx < total; idx += stride) {
        float r = 1.0f + dists[idx] / sigma;
        kout[idx] = 1.0f / sqrtf(r);
    }
}

// ---------------- s[j] = sum_i 0.5*k[j,i]^3   (W row sums, W symmetric) ------------
__global__ __launch_bounds__(256) void row_sums(const float* __restrict__ kmat,
                                                float* __restrict__ s) {
    int j = blockIdx.x * blockDim.x + threadIdx.x;   // one thread per row
    const float* row = kmat + (size_t)j * NPTS;
    float acc = 0.0f;
    #pragma unroll 4
    for (int i = 0; i < NPTS; i += 4) {
        v4f v = *(const v4f*)(row + i);
        acc += 0.5f * v.x * v.x * v.x;
        acc += 0.5f * v.y * v.y * v.y;
        acc += 0.5f * v.z * v.z * v.z;
        acc += 0.5f * v.w * v.w * v.w;
    }
    s[j] = acc;
}

// ---------------- der = (2/sigma) * (W X - s .* X), W[i,j] = 0.5*k^3 ----------------
// GEMM (1024x1024)x(1024x512) on fp32 WMMA; A fragment computed on the fly from k.
__global__ __launch_bounds__(256) void wmma_der(const float* __restrict__ kmat,
                                                const float* __restrict__ X,
                                                const float* __restrict__ s,
                                                const float* __restrict__ state_f,
                                                float* __restrict__ der) {
    int wave = blockIdx.x * 8 + (threadIdx.x >> 5);  // 2048 tiles: 64 (j) x 32 (d)
    int lane = threadIdx.x & 31;
    int tj = wave >> 5;
    int td = wave & 31;
    int j0 = tj * 16, d0 = td * 16;

    int m  = lane & 15;
    int kb = (lane >> 4) * 2;

    const float* aptr = kmat + (size_t)(j0 + m) * NPTS + kb;   // W row j0+m
    const float* bptr = X + (size_t)kb * DIM + d0 + m;         // X[k][d0+n], n = lane&15

    float tos = state_f[3];   // 2/sigma

    v8f acc = {};
    #pragma unroll 8
    for (int k0 = 0; k0 < NPTS; k0 += 4) {
        v2f kk = *(const v2f*)(aptr + k0);
        v2f a;
        a.x = 0.5f * kk.x * kk.x * kk.x;
        a.y = 0.5f * kk.y * kk.y * kk.y;
        v2f b;
        b.x = bptr[(size_t)k0 * DIM];
        b.y = bptr[(size_t)k0 * DIM + DIM];
        acc = __builtin_amdgcn_wmma_f32_16x16x4_f32(
            false, a, false, b, (short)0, acc, false, false);
    }

    int n  = lane & 15;
    int hi = (lane >> 4) * 8;
    #pragma unroll
    for (int v = 0; v < 8; ++v) {
        int j = j0 + hi + v;
        float xv = X[(size_t)j * DIM + d0 + n];
        der[(size_t)j * DIM + d0 + n] = (acc[v] - s[j] * xv) * tos;
    }
}

extern "C" void kernel_launch(void* const* d_in, const int* in_sizes, int n_in,
                              void* d_out, int out_size, void* d_ws, size_t ws_size,
                              hipStream_t stream) {
    const float* X   = (const float*)d_in[0];
    const float* EMA = (const float*)d_in[1];

    float* out    = (float*)d_out;
    float* kout   = out;                                       // [N,N]
    float* derout = out + (size_t)NPTS * NPTS;                 // [N,D]
    float* sigout = out + (size_t)NPTS * NPTS + (size_t)NPTS * DIM;  // scalar

    float*    ws    = (float*)d_ws;
    float*    dists = ws;                                      // N*N floats (4 MB)
    float*    sq    = ws + (size_t)NPTS * NPTS;                // N floats
    float*    s     = sq + NPTS;                               // N floats
    unsigned* hist  = (unsigned*)(s + NPTS);                   // 256 u32
    unsigned* state = hist + 256;                              // prefix, krem, sigma, 2/sigma
    float*    state_f = (float*)state;

    init_state<<<1, 256, 0, stream>>>(hist, state);
    row_sqnorm<<<NPTS / 256, 256, 0, stream>>>(X, sq);
    wmma_gram_dists<<<512, 256, 0, stream>>>(X, sq, dists);    // 4096 waves, 16x16 tiles

    for (int p = 0; p < 4; ++p) {
        int shift = 24 - 8 * p;
        unsigned mask = (p == 0) ? 0u : (0xFFFFFFFFu << (shift + 8));
        hist_pass<<<512, 256, 0, stream>>>(dists, hist, state, shift, mask);
        radix_select<<<1, 1, 0, stream>>>(hist, state, shift);
    }
    finalize_sigma<<<1, 1, 0, stream>>>(EMA, state, sigout);

    compute_k<<<1024, 256, 0, stream>>>(dists, state_f, kout);
    row_sums<<<NPTS / 256, 256, 0, stream>>>(kout, s);
    wmma_der<<<256, 256, 0, stream>>>(kout, X, s, state_f, derout);
}